// TravelGNN_83184926589431
// MI455X (gfx1250) — compile-verified
//
#include <hip/hip_runtime.h>

typedef __attribute__((ext_vector_type(2))) float v2f;
typedef __attribute__((ext_vector_type(8))) float v8f;

// ---------- helpers ----------
__device__ __forceinline__ float lrelu02(float a) { return a > 0.f ? a : 0.2f * a; }

// order-preserving float<->uint encoding for atomicMax on floats
__device__ __forceinline__ unsigned enc_ord(float f) {
  unsigned u = __float_as_uint(f);
  return (u & 0x80000000u) ? ~u : (u | 0x80000000u);
}
__device__ __forceinline__ float dec_ord(unsigned u) {
  u = (u & 0x80000000u) ? (u & 0x7FFFFFFFu) : ~u;
  return __uint_as_float(u);
}

// Guaranteed-hardware float atomic add: GLOBAL_ATOMIC_ADD_F32, non-returning
// (STOREcnt-tracked; s_endpgm's implicit wait-idle orders it vs. later kernels).
__device__ __forceinline__ void atomic_add_f32(float* p, float v) {
#if defined(__gfx1250__)
  asm volatile("global_atomic_add_f32 %0, %1, off" : : "v"(p), "v"(v) : "memory");
#else
  atomicAdd(p, v);
#endif
}

// ---------- dense transform: H[N x 64] = X[N x K] @ W[K x 64], fp32 WMMA ----------
// One wave computes a 16x16 tile via V_WMMA_F32_16X16X4_F32, K stepped by 4.
// A layout (16x4): lane l -> row m0+(l&15), Ks k+2*(l>>4)+{0,1} in vgpr{0,1}
// B layout (4x16): lane l -> col n0+(l&15), Ks k+2*(l>>4)+{0,1} in vgpr{0,1}
// C/D layout    : acc[r] = H[m0 + r + 8*(l>>4)][n0 + (l&15)]
__global__ void __launch_bounds__(256)
gemm_xw(const float* __restrict__ X, const float* __restrict__ W,
        float* __restrict__ H, int N, int K) {
  const int lane = threadIdx.x & 31;
  const int wave = threadIdx.x >> 5;           // 0..7
  const int half = lane >> 4;                  // 0/1
  const int l15  = lane & 15;
  const int m0 = blockIdx.x * 32 + (wave >> 2) * 16;
  const int n0 = (wave & 3) * 16;
  if (m0 >= N) return;                         // wave-uniform: EXEC stays all-ones
  v8f acc = {};
#if __has_builtin(__builtin_amdgcn_wmma_f32_16x16x4_f32)
  const float* xrow = X + (size_t)(m0 + l15) * K + 2 * half;
  const float* wcol = W + (size_t)(2 * half) * 64 + n0 + l15;
  for (int k = 0; k < K; k += 4) {
    v2f a = { xrow[k], xrow[k + 1] };
    v2f b = { wcol[(size_t)k * 64], wcol[(size_t)(k + 1) * 64] };
    acc = __builtin_amdgcn_wmma_f32_16x16x4_f32(
        false, a, false, b, (short)0, acc, false, false);
  }
#else
  // scalar fallback (also used for host-pass parse)
  for (int k = 0; k < K; ++k) {
    float b = W[(size_t)k * 64 + n0 + l15];
#pragma unroll
    for (int r = 0; r < 8; ++r)
      acc[r] += X[(size_t)(m0 + r + 8 * half) * K + k] * b;
  }
#endif
  float* hrow = H + (size_t)(m0 + 8 * half) * 64 + n0 + l15;
#pragma unroll
  for (int r = 0; r < 8; ++r) hrow[(size_t)r * 64] = acc[r];
}

// ---------- wa[K] = W[K x 64] @ a[64] (tiny) ----------
__global__ void wvec(const float* __restrict__ W, const float* __restrict__ a,
                     float* __restrict__ wa, int K) {
  int k = threadIdx.x;
  if (k < K) {
    float s = 0.f;
#pragma unroll 8
    for (int j = 0; j < 64; ++j) s += W[(size_t)k * 64 + j] * a[j];
    wa[k] = s;
  }
}

// ---------- alpha[n] = dot(X[n,:], wa) ----------
__global__ void node_alpha(const float* __restrict__ X, const float* __restrict__ wa,
                           float* __restrict__ alpha, int N, int K) {
  int n = blockIdx.x * blockDim.x + threadIdx.x;
  if (n >= N) return;
  const float* row = X + (size_t)n * K;
  float s = 0.f;
  for (int k = 0; k < K; ++k) s += row[k] * wa[k];
  alpha[n] = s;
}

// ---------- fills / bias init ----------
__global__ void fill_u32(unsigned* __restrict__ p, unsigned v, int n) {
  int i = blockIdx.x * blockDim.x + threadIdx.x;
  if (i < n) p[i] = v;
}
__global__ void fill_f32(float* __restrict__ p, float v, int n) {
  int i = blockIdx.x * blockDim.x + threadIdx.x;
  if (i < n) p[i] = v;
}
__global__ void init_bias(float* __restrict__ out, size_t n,
                          const float* b0, const float* b1, const float* b2,
                          const float* b3, const float* b4) {
  size_t i = (size_t)blockIdx.x * blockDim.x + threadIdx.x;
  if (i >= n) return;
  int c = (int)(i & 63);
  float s = b0 ? b0[c] : 0.f;
  if (b1) s += b1[c];
  if (b2) s += b2[c];
  if (b3) s += b3[c];
  if (b4) s += b4[c];
  out[i] = s;
}

// ---------- edge pass 1: segment max of leaky(alpha_s[src]+alpha_d[dst]) ----------
__global__ void edge_max(const int* __restrict__ src, const int* __restrict__ dst,
                         const float* __restrict__ as_, const float* __restrict__ ad_,
                         unsigned* __restrict__ menc, int E) {
  int e = blockIdx.x * blockDim.x + threadIdx.x;
  if (e >= E) return;
  int d = dst[e];
  float a = lrelu02(as_[src[e]] + ad_[d]);
  atomicMax(&menc[d], enc_ord(a));
}

// ---------- edge pass 2: segment sum of exp(a - m[dst]) ----------
__global__ void edge_expsum(const int* __restrict__ src, const int* __restrict__ dst,
                            const float* __restrict__ as_, const float* __restrict__ ad_,
                            const unsigned* __restrict__ menc, float* __restrict__ esum,
                            int E) {
  int e = blockIdx.x * blockDim.x + threadIdx.x;
  if (e >= E) return;
  int d = dst[e];
  float a = lrelu02(as_[src[e]] + ad_[d]);
  atomic_add_f32(&esum[d], expf(a - dec_ord(menc[d])));
}

// ---------- edge pass 3: out[dst,:] += w_e * hs[src,:]  (one wave per edge) ----------
__global__ void __launch_bounds__(256)
edge_aggregate(const int* __restrict__ src, const int* __restrict__ dst,
               const float* __restrict__ as_, const float* __restrict__ ad_,
               const unsigned* __restrict__ menc, const float* __restrict__ esum,
               const float* __restrict__ hs, float* __restrict__ out, int E) {
  const int lane = threadIdx.x & 31;
  long wid = (long)blockIdx.x * (blockDim.x >> 5) + (threadIdx.x >> 5);
  long nw  = (long)gridDim.x * (blockDim.x >> 5);
  for (long e = wid; e < E; e += nw) {
    int s = src[e], d = dst[e];
    float a = lrelu02(as_[s] + ad_[d]);            // wave-uniform scalar loads
    float w = expf(a - dec_ord(menc[d])) / esum[d];
    const float* hrow = hs + (size_t)s * 64;
    float* orow = out + (size_t)d * 64;
    atomic_add_f32(&orow[lane],      w * hrow[lane]);
    atomic_add_f32(&orow[lane + 32], w * hrow[lane + 32]);
  }
}

// ---------- final ReLU ----------
__global__ void relu_inplace(float* __restrict__ p, size_t n) {
  size_t i = (size_t)blockIdx.x * blockDim.x + threadIdx.x;
  if (i < n) p[i] = fmaxf(p[i], 0.f);
}

// ===================== host launcher =====================
extern "C" void kernel_launch(void* const* d_in, const int* in_sizes, int n_in,
                              void* d_out, int out_size, void* d_ws, size_t ws_size,
                              hipStream_t stream) {
  (void)n_in; (void)ws_size;
  // node types: 0 person(K=64) 1 household(32) 2 zone(16) 3 purpose(16)
  const int KT[4] = {64, 32, 16, 16};
  const float* X[4];
  long NT[4];
  for (int t = 0; t < 4; ++t) {
    X[t] = (const float*)d_in[t];
    NT[t] = (long)in_sizes[t] / KT[t];
  }
  size_t off[4];
  off[0] = 0;
  off[1] = off[0] + (size_t)NT[0] * 64;
  off[2] = off[1] + (size_t)NT[1] * 64;
  off[3] = off[2] + (size_t)NT[2] * 64;

  float* out = (float*)d_out;
  float* ws  = (float*)d_ws;
  const size_t NP = (size_t)NT[0];                 // max node count (person)
  float*    hs   = ws;                             // NP*64
  float*    asv  = hs + NP * 64;                   // NP
  float*    adv  = asv + NP;                       // NP
  unsigned* menc = (unsigned*)(adv + NP);          // NP
  float*    esum = (float*)(menc + NP);            // NP
  float*    waS  = esum + NP;                      // 64
  float*    waD  = waS + 64;                       // 64

  // relations: {src_type, dst_type, src_idx, dst_idx, params_base}
  // params flattened in RELS insertion order: Ws, Wd, as, ad, b per relation, base 22
  struct RelDesc { int st, dt, si, di, pb; };
  const RelDesc R[9] = {
    {0, 1,  4,  5, 22 + 0 * 5},  // belongs_to person->household
    {1, 2,  6,  7, 22 + 1 * 5},  // located_in household->zone
    {2, 3,  8,  9, 22 + 2 * 5},  // has_purpose zone->purpose
    {0, 3, 10, 11, 22 + 3 * 5},  // performs person->purpose
    {0, 0, 12, 13, 22 + 4 * 5},  // parent
    {0, 0, 14, 15, 22 + 5 * 5},  // child
    {0, 0, 16, 17, 22 + 6 * 5},  // spouse
    {0, 0, 18, 19, 22 + 7 * 5},  // housemate
    {0, 0, 20, 21, 22 + 8 * 5},  // sibling
  };

  // per-destination-type bias sum written into d_out first
  const float* bp[4][5] = {{0}};
  int bc[4] = {0, 0, 0, 0};
  for (int r = 0; r < 9; ++r) {
    int dt = R[r].dt;
    bp[dt][bc[dt]++] = (const float*)d_in[R[r].pb + 4];
  }
  for (int t = 0; t < 4; ++t) {
    size_t n = (size_t)NT[t] * 64;
    unsigned blocks = (unsigned)((n + 255) / 256);
    init_bias<<<blocks, 256, 0, stream>>>(out + off[t], n,
        bp[t][0], bp[t][1], bp[t][2], bp[t][3], bp[t][4]);
  }

  for (int r = 0; r < 9; ++r) {
    const RelDesc& rd = R[r];
    const float* xs = X[rd.st];
    const float* xd = X[rd.dt];
    const int Ks = KT[rd.st], Kd = KT[rd.dt];
    const int Ns = (int)NT[rd.st], Nd = (int)NT[rd.dt];
    const int* src = (const int*)d_in[rd.si];
    const int* dst = (const int*)d_in[rd.di];
    const int E = in_sizes[rd.si];
    const float* Wsrc = (const float*)d_in[rd.pb + 0];
    const float* Wdst = (const float*)d_in[rd.pb + 1];
    const float* avs  = (const float*)d_in[rd.pb + 2];
    const float* avd  = (const float*)d_in[rd.pb + 3];
    float* outD = out + off[rd.dt];

    // fold attention vectors through the weights: alpha = x @ (W @ a)
    wvec<<<1, 64, 0, stream>>>(Wsrc, avs, waS, Ks);
    wvec<<<1, 64, 0, stream>>>(Wdst, avd, waD, Kd);
    // hs = x_s @ Ws  (WMMA)
    gemm_xw<<<(unsigned)((Ns + 31) / 32), 256, 0, stream>>>(xs, Wsrc, hs, Ns, Ks);
    // per-node attention scalars
    node_alpha<<<(unsigned)((Ns + 255) / 256), 256, 0, stream>>>(xs, waS, asv, Ns, Ks);
    node_alpha<<<(unsigned)((Nd + 255) / 256), 256, 0, stream>>>(xd, waD, adv, Nd, Kd);
    // segment softmax state: m = -inf (encoded), esum = 0
    fill_u32<<<(unsigned)((Nd + 255) / 256), 256, 0, stream>>>(menc, 0x007FFFFFu, Nd);
    fill_f32<<<(unsigned)((Nd + 255) / 256), 256, 0, stream>>>(esum, 0.f, Nd);
    // three edge passes
    edge_max<<<(unsigned)((E + 255) / 256), 256, 0, stream>>>(src, dst, asv, adv, menc, E);
    edge_expsum<<<(unsigned)((E + 255) / 256), 256, 0, stream>>>(src, dst, asv, adv, menc, esum, E);
    edge_aggregate<<<(unsigned)((E + 7) / 8), 256, 0, stream>>>(src, dst, asv, adv,
                                                                menc, esum, hs, outD, E);
  }

  size_t n = (size_t)out_size;
  relu_inplace<<<(unsigned)((n + 255) / 256), 256, 0, stream>>>(out, n);
}